// DecorrLoss_20340965113926
// MI455X (gfx1250) — compile-verified
//
#include <hip/hip_runtime.h>

typedef float v2f __attribute__((ext_vector_type(2)));
typedef float v4f __attribute__((ext_vector_type(4)));
typedef float v8f __attribute__((ext_vector_type(8)));
typedef unsigned int u32x4 __attribute__((ext_vector_type(4)));
typedef int i32x4 __attribute__((ext_vector_type(4)));
typedef int i32x8 __attribute__((ext_vector_type(8)));

#define DDIM      512
#define NROWS     32768            // 8 * 4096
#define KC        32               // k rows staged in LDS per chunk
#define KSPLIT    16
#define NCHUNK    ((NROWS / KSPLIT) / KC)   // 64 chunks per split
#define XA_STRIDE 80               // 64 cols + 16 pad dwords (TDM pad_interval=64)
#define XB_STRIDE 144              // 128 cols + 16 pad dwords (TDM pad_interval=128)
#define XA_SZ     (KC * XA_STRIDE) // floats
#define XB_SZ     (KC * XB_STRIDE) // floats
#define BUF_SZ    (XA_SZ + XB_SZ)  // floats, one double-buffer slot
#define NBLK      20               // 64x128 blocks touching the upper triangle

// ---------------------------------------------------------------------------
// ws layout: [0, 512*512) Gram accumulator S = sum_n x_r x_c (kept blocks only)
//            [512*512]    sum_n (row_sq^2 - quart)
//            [512*512+1]  sum_n (quart - 2*row_sq + d)
// ---------------------------------------------------------------------------

__global__ void zero_ws(float* ws) {
    const int total = DDIM * DDIM + 2;
    for (int i = blockIdx.x * blockDim.x + threadIdx.x; i < total;
         i += gridDim.x * blockDim.x)
        ws[i] = 0.0f;
}

// One wave (32 lanes) per row: row_sq = sum x^2, quart = sum x^4.
__global__ __launch_bounds__(256) void loss_kernel(const float* __restrict__ x,
                                                   float* __restrict__ ws) {
    const int lane = threadIdx.x & 31;
    const int wave = threadIdx.x >> 5;
    const int row  = blockIdx.x * 8 + wave;     // gridDim.x = 4096 -> 32768 rows
    const v4f* xr = (const v4f*)(x + (size_t)row * DDIM);
    float s2 = 0.0f, s4 = 0.0f;
#pragma unroll
    for (int q = 0; q < 4; ++q) {
        v4f v = xr[lane + 32 * q];
#pragma unroll
        for (int e = 0; e < 4; ++e) {
            float t2 = v[e] * v[e];
            s2 += t2;
            s4 += t2 * t2;
        }
    }
#pragma unroll
    for (int off = 16; off > 0; off >>= 1) {
        s2 += __shfl_xor(s2, off, 32);
        s4 += __shfl_xor(s4, off, 32);
    }
    if (lane == 0) {
        atomicAdd(&ws[DDIM * DDIM],     s2 * s2 - s4);
        atomicAdd(&ws[DDIM * DDIM + 1], s4 - 2.0f * s2 + (float)DDIM);
    }
}

// TDM: async-load a KC x tile_w dword tile (row stride 512 dwords in memory)
// into LDS at byte offset lds_off, inserting 16 dwords of pad every
// 2^(pad_iv+1) dwords (bank-conflict-free layout, no shuffle cost).
__device__ __forceinline__ void tdm_load_tile(unsigned lds_off,
                                              const float* gptr,
                                              unsigned tile_w, unsigned pad_iv) {
    const unsigned long long ga = (unsigned long long)(size_t)gptr;
    u32x4 g0;
    g0[0] = 1u;                                   // count=1 (valid), user mode
    g0[1] = lds_off;                              // lds_addr (bytes)
    g0[2] = (unsigned)ga;                         // global_addr[31:0]
    g0[3] = (unsigned)(ga >> 32) | (2u << 30);    // global_addr[56:32] | type=2
    i32x8 g1;
    g1[0] = (int)((2u << 16)        // data_size = 4 bytes
                | (1u << 20)        // pad_enable
                | (pad_iv << 22)    // pad_interval: 5 -> 64 dw, 6 -> 128 dw
                | (15u << 25));     // pad_amount: 15 -> 16 dwords
    g1[1] = (int)(512u << 16);      // tensor_dim0 = 512 (bits 79:48 low half)
    g1[2] = (int)(0x8000u << 16);   // tensor_dim1 = 32768 (bits 111:80 low half)
    g1[3] = (int)(tile_w << 16);    // tile_dim0 (bits 127:112)
    g1[4] = (int)KC;                // tile_dim1 = 32, tile_dim2 = 0
    g1[5] = 512;                    // tensor_dim0_stride low
    g1[6] = 0;
    g1[7] = 0;
    i32x4 z = {0, 0, 0, 0};
#if defined(__clang_major__) && (__clang_major__ >= 23)
    i32x8 z8 = {0, 0, 0, 0, 0, 0, 0, 0};
    __builtin_amdgcn_tensor_load_to_lds(g0, g1, z, z, z8, 0);
#else
    __builtin_amdgcn_tensor_load_to_lds(g0, g1, z, z, 0);
#endif
}

// Gram via V_WMMA_F32_16X16X4_F32. Workgroup = 64x128 output block
// (4x8 tiles, 8 waves x 4 accumulator tiles), TDM double-buffered LDS
// staging, split-K over samples, fp32 atomics fold the splits.
__global__ __launch_bounds__(256) void gram_kernel(const float* __restrict__ x,
                                                   float* __restrict__ S) {
    __shared__ float lds[2 * BUF_SZ];           // ~57 KB

    // decode (bi, bj2): 64-row block bi (0..7), 128-col block bj2 (0..3),
    // kept iff it touches the upper triangle: 2*bj2 + 1 >= bi.
    int pair  = blockIdx.x % NBLK;
    int split = blockIdx.x / NBLK;
    int bi = 0, u = pair, bj2 = 0;
    for (;;) {
        const int mn  = (bi <= 1) ? 0 : (((bi - 2) >> 1) + 1);
        const int cnt = 4 - mn;
        if (u < cnt) { bj2 = mn + u; break; }
        u -= cnt;
        ++bi;
    }
    const int Ibase = bi * 64;
    const int Jbase = bj2 * 128;

    const int tid  = threadIdx.x;
    const int lane = tid & 31;
    const int w    = tid >> 5;
    const int ti   = w & 3;                     // tile row 0..3
    const int tjb  = (w >> 2) * 4;              // tile cols tjb..tjb+3

    // WMMA fragment addressing: lane L -> M/N = L%16, K = 2*(L/16)+v
    const int fr = 2 * (lane >> 4);
    const int fn = lane & 15;

    v8f acc[4] = {v8f{}, v8f{}, v8f{}, v8f{}};

    const unsigned lds_base = (unsigned)(size_t)&lds[0];
    const int k0beg = split * (NROWS / KSPLIT);

    if (w == 0) {   // prologue: chunk 0 into buffer 0
        tdm_load_tile(lds_base,             x + (size_t)k0beg * DDIM + Ibase,  64, 5);
        tdm_load_tile(lds_base + XA_SZ * 4, x + (size_t)k0beg * DDIM + Jbase, 128, 6);
    }

    for (int c = 0; c < NCHUNK; ++c) {
        if (w == 0) {
            if (c + 1 < NCHUNK) {           // prefetch next chunk into other buffer
                const int k1 = k0beg + (c + 1) * KC;
                const unsigned bo = lds_base + ((c + 1) & 1) * (BUF_SZ * 4);
                tdm_load_tile(bo,             x + (size_t)k1 * DDIM + Ibase,  64, 5);
                tdm_load_tile(bo + XA_SZ * 4, x + (size_t)k1 * DDIM + Jbase, 128, 6);
                __builtin_amdgcn_s_wait_tensorcnt(2);   // current chunk landed
            } else {
                __builtin_amdgcn_s_wait_tensorcnt(0);
            }
        }
        __syncthreads();

        const float* Xa = lds + (c & 1) * BUF_SZ;
        const float* Xb = Xa + XA_SZ;
#pragma unroll 2
        for (int kk = 0; kk < KC; kk += 4) {
            const int ra = (kk + fr) * XA_STRIDE;
            const int rb = (kk + fr) * XB_STRIDE;
            v2f a;
            a[0] = Xa[ra +             16 * ti + fn];
            a[1] = Xa[ra + XA_STRIDE + 16 * ti + fn];
#pragma unroll
            for (int t = 0; t < 4; ++t) {
                v2f b;
                b[0] = Xb[rb +             16 * (tjb + t) + fn];
                b[1] = Xb[rb + XB_STRIDE + 16 * (tjb + t) + fn];
                acc[t] = __builtin_amdgcn_wmma_f32_16x16x4_f32(
                    false, a, false, b, (short)0, acc[t], false, false);
            }
        }
        __syncthreads();
    }

    // C/D layout: lane L, VGPR v -> M = v + 8*(L/16), N = L%16
    const int row0 = Ibase + 16 * ti + 8 * (lane >> 4);
#pragma unroll
    for (int t = 0; t < 4; ++t) {
        const int col = Jbase + 16 * (tjb + t) + fn;
#pragma unroll
        for (int v = 0; v < 8; ++v)
            atomicAdd(&S[(size_t)(row0 + v) * DDIM + col], acc[t][v]);
    }
}

__global__ void finalize(const float* __restrict__ ws,
                         const float* __restrict__ kappa_p,
                         float* __restrict__ out) {
    const int idx = blockIdx.x * blockDim.x + threadIdx.x;
    const float kappa = *kappa_p;
    if (idx < DDIM * DDIM) {
        const int r = idx >> 9, c = idx & (DDIM - 1);
        // block (r>>6, c>>7) was computed iff 2*(c>>7)+1 >= (r>>6)
        const bool direct = (2 * (c >> 7) + 1) >= (r >> 6);
        const float s = direct ? ws[idx] : ws[(size_t)c * DDIM + r];
        const float g = s * (1.0f / (float)NROWS);
        out[idx] = (r == c) ? kappa * (g - 1.0f) : (1.0f - kappa) * g;
    }
    if (idx == 0) {
        const float inv = 1.0f / ((float)NROWS * (float)DDIM * (float)DDIM);
        out[DDIM * DDIM]     = ws[DDIM * DDIM] * inv;
        out[DDIM * DDIM + 1] = ws[DDIM * DDIM + 1] * inv;
    }
}

extern "C" void kernel_launch(void* const* d_in, const int* in_sizes, int n_in,
                              void* d_out, int out_size, void* d_ws, size_t ws_size,
                              hipStream_t stream) {
    const float* x     = (const float*)d_in[0];
    const float* kappa = (const float*)d_in[1];
    float* out = (float*)d_out;
    float* ws  = (float*)d_ws;

    zero_ws<<<512, 256, 0, stream>>>(ws);
    loss_kernel<<<NROWS / 8, 256, 0, stream>>>(x, ws);
    gram_kernel<<<NBLK * KSPLIT, 256, 0, stream>>>(x, ws);
    finalize<<<(DDIM * DDIM + 255) / 256, 256, 0, stream>>>(ws, kappa, out);
}